// RoiPoolingLayer_38534446579827
// MI455X (gfx1250) — compile-verified
//
#include <hip/hip_runtime.h>

// Problem constants (fixed by the reference).
constexpr int NB = 16;    // batch
constexpr int NH = 100;   // feature map height
constexpr int NW = 100;   // feature map width
constexpr int NC = 256;   // channels (NHWC innermost, contiguous)
constexpr int NR = 128;   // rois per batch
constexpr int PP = 7;     // pool size

// Per-stage staging: 2 source rows x 14 corner columns, each column = NC
// contiguous floats (1 KB). 28 segments * 1 KB = 28 KB per buffer.
constexpr int SEGS      = 2 * 2 * PP;        // 28
constexpr int BUF_FLOATS = SEGS * NC;        // 7168 floats = 28 KB
constexpr int BUF_BYTES  = BUF_FLOATS * 4;   // 28672

__device__ __forceinline__ void wait_async0() {
#if __has_builtin(__builtin_amdgcn_s_wait_asynccnt)
    __builtin_amdgcn_s_wait_asynccnt(0);
#else
    asm volatile("s_wait_asynccnt 0" ::: "memory");
#endif
}

// Per-lane async copy of 16 bytes global -> LDS (GLOBAL_LOAD_ASYNC_TO_LDS_B128,
// GV mode: 64-bit VGPR address pair, LDS byte offset in a VGPR; ASYNCcnt).
__device__ __forceinline__ void async_copy_b128(unsigned lds_off, const float* gaddr) {
    asm volatile("global_load_async_to_lds_b128 %0, %1, off"
                 :: "v"(lds_off), "v"(gaddr)
                 : "memory");
}

__global__ __launch_bounds__(256) void
roi_align_async_kernel(const float* __restrict__ fm,
                       const int*   __restrict__ rois,
                       float*       __restrict__ out)
{
    __shared__ __align__(128) float buf[2][BUF_FLOATS];
    __shared__ int   sx0[PP], sx1[PP], sy0[PP], sy1[PP];
    __shared__ float sfx[PP], sfy[PP];

    const int t   = threadIdx.x;          // 0..255 == channel id in compute phase
    const int blk = blockIdx.x;           // one ROI per block: blk = b*NR + r
    const int b   = blk / NR;

    // roi = {x, y, h, w}
    const int4 roi = reinterpret_cast<const int4*>(rois)[blk];

    // --- Setup: TF2 half-pixel-center bilinear source coordinates ------------
    if (t < PP) {
        {   // x axis uses size = w (roi.w)
            float size = (float)roi.w;
            float src  = ((float)t + 0.5f) * (1.0f / (float)PP) * size - 0.5f;
            src = fminf(fmaxf(src, 0.0f), size - 1.0f);
            int i0 = (int)floorf(src);
            sfx[t] = src - (float)i0;
            int g0 = i0 + roi.x;
            sx0[t] = g0;
            sx1[t] = min(g0 + 1, roi.x + roi.w - 1);
        }
        {   // y axis uses size = h (roi.z)
            float size = (float)roi.z;
            float src  = ((float)t + 0.5f) * (1.0f / (float)PP) * size - 0.5f;
            src = fminf(fmaxf(src, 0.0f), size - 1.0f);
            int j0 = (int)floorf(src);
            sfy[t] = src - (float)j0;
            int g0 = j0 + roi.y;
            sy0[t] = g0;
            sy1[t] = min(g0 + 1, roi.y + roi.z - 1);
        }
    }
    __syncthreads();

    const float* fmb = fm + (size_t)b * NH * NW * NC;

    // --- Hoist per-thread issue addressing (stage-invariant) ------------------
    // Each thread owns 7 of the 1792 16B lane-copies per stage.
    // Segment s = r*14 + j: r selects which source row (y0 vs y1),
    // j = 2*px + (0:x0, 1:x1) selects the corner column.
    const float* colbase[7];   // fmb + col*NC + off*4   (row term added per stage)
    unsigned     lds0[7];      // LDS byte offset within buf[0]
    int          rsel[7];      // 0 -> y0 row, 1 -> y1 row
#pragma unroll
    for (int i = 0; i < 7; ++i) {
        int idx = i * 256 + t;            // 0..1791
        int seg = idx >> 6;               // 0..27  (64 x 16B per segment)
        int off = idx & 63;               // 16-byte unit within segment
        int r   = seg / 14;
        int j   = seg - r * 14;
        int px  = j >> 1;
        int col = (j & 1) ? sx1[px] : sx0[px];
        rsel[i]    = r;
        colbase[i] = fmb + (size_t)col * NC + off * 4;
        lds0[i]    = (unsigned)(size_t)&buf[0][seg * NC + off * 4];
    }

    // Preload the 7 horizontal lerp weights into registers.
    float fxr[PP];
#pragma unroll
    for (int px = 0; px < PP; ++px) fxr[px] = sfx[px];

    auto issue_stage = [&](int py, int sel) {
        const int ya = sy0[py];
        const int yb = sy1[py];
        const unsigned ldsadd = (unsigned)sel * BUF_BYTES;
#pragma unroll
        for (int i = 0; i < 7; ++i) {
            const int row = rsel[i] ? yb : ya;
            async_copy_b128(lds0[i] + ldsadd,
                            colbase[i] + (size_t)row * (NW * NC));
        }
    };

    // Prime the pipeline, then double-buffer: while computing row `py` from
    // buf[sel], the async engine fills buf[sel^1] with row `py+1`.
    issue_stage(0, 0);

    for (int py = 0; py < PP; ++py) {
        const int sel = py & 1;

        wait_async0();        // my wave's in-flight async copies are done
        __syncthreads();      // all waves' copies landed + prev compute finished

        if (py < PP - 1) issue_stage(py + 1, sel ^ 1);

        const float wy = sfy[py];
        float* orow = out + (((size_t)blk * PP + py) * PP) * NC + t;
#pragma unroll
        for (int px = 0; px < PP; ++px) {
            const float wx = fxr[px];
            const float a  = buf[sel][(2 * px         ) * NC + t]; // (y0, x0)
            const float c1 = buf[sel][(2 * px + 1     ) * NC + t]; // (y0, x1)
            const float c2 = buf[sel][(14 + 2 * px    ) * NC + t]; // (y1, x0)
            const float d  = buf[sel][(14 + 2 * px + 1) * NC + t]; // (y1, x1)
            const float top = a  + (c1 - a ) * wx;
            const float bot = c2 + (d  - c2) * wx;
            // Write-once output: non-temporal store keeps the feature map
            // resident in the 192 MB L2 instead of being evicted by output.
            __builtin_nontemporal_store(top + (bot - top) * wy,
                                        &orow[(size_t)px * NC]);
        }
    }
}

extern "C" void kernel_launch(void* const* d_in, const int* in_sizes, int n_in,
                              void* d_out, int out_size, void* d_ws, size_t ws_size,
                              hipStream_t stream) {
    (void)in_sizes; (void)n_in; (void)d_ws; (void)ws_size; (void)out_size;
    const float* fm   = (const float*)d_in[0];
    const int*   rois = (const int*)d_in[1];
    float*       out  = (float*)d_out;

    dim3 grid(NB * NR);   // one block per ROI (2048 blocks, 8 wave32 waves each)
    dim3 block(256);      // thread == channel in the compute phase
    roi_align_async_kernel<<<grid, block, 0, stream>>>(fm, rois, out);
}